// SpatiallySparseLocallyConnected3D_2087354106281
// MI455X (gfx1250) — compile-verified
//
#include <hip/hip_runtime.h>

// Locally-connected 3D conv (unshared weights), valid padding, stride 1.
// x:      (B=2, 24,24,24, C=8)        fp32
// kernel: (P=10648, K=216, F=8)       fp32
// bias:   (22,22,22, F=8)             fp32
// out:    (B=2, 22,22,22, F=8)        fp32
//
// Strategy (MI455X, bandwidth-bound @ ~73.6MB weights / 23.3 TB/s):
// one wave32 per spatial site; both batches fused so every weight element is
// read exactly once; per K-chunk-of-4 one V_WMMA_F32_16X16X4_F32 with
//   A (16x4):  rows 0..1 = batches (patch values), other rows = don't-care
//   B (4x16):  cols 0..7 = output features (weights),  cols 8..15 = don't-care
//   D (16x16): row 0 -> out[b=0], row 1 -> out[b=1]

typedef __attribute__((ext_vector_type(2))) float v2f;
typedef __attribute__((ext_vector_type(8))) float v8f;

#define OR_      22
#define P_TOTAL  (22*22*22)       /* 10648 */
#define F_OUT    8
#define K_TOT    216              /* 27 offsets * 8 channels */
#define KF       (K_TOT*F_OUT)    /* 1728 floats per site's weight slice */
/* x strides in floats: (B, D, H, W, C) = (2,24,24,24,8) */
#define XS_B     (24*24*24*8)     /* 110592 */
#define XS_D     (24*24*8)        /* 4608 */
#define XS_H     (24*8)           /* 192 */
#define XS_W     8

__global__ __launch_bounds__(256) void lc3d_wmma_f32_kernel(
    const float* __restrict__ x,
    const float* __restrict__ kern,
    const float* __restrict__ bias,
    float* __restrict__ out)
{
    const int lane = threadIdx.x & 31;
    const int wave = threadIdx.x >> 5;
    const int p    = blockIdx.x * 8 + wave;      // grid sized exactly: p < 10648

    // p -> (i,j,z) site coordinates
    const int i   = p / (OR_ * OR_);
    const int rem = p - i * (OR_ * OR_);
    const int j   = rem / OR_;
    const int z   = rem - j * OR_;

    const int half = lane >> 4;     // K-split halves of the wave
    const int lrow = lane & 15;     // A: matrix row (M); B: matrix column (N)

    // A-fragment source: lane lrow==0 -> batch 0, lrow>=1 -> batch 1
    // (lanes 2..15 duplicate batch-1's address; their garbage feeds unused D rows)
    const int bsel = (lrow >= 1) ? 1 : 0;
    const float* xv_base =
        x + bsel * XS_B + i * XS_D + j * XS_H + z * XS_W + 2 * half;

    // B-fragment source: column n = lrow&7 (cols 8..15 duplicate -> unused D cols)
    const float* kslice = kern + (size_t)p * KF;
    const float* kp     = kslice + (lrow & 7);

    // Fan the 6912-byte weight slice across lanes: one global_prefetch_b8 that
    // touches the whole slice region ahead of the load loop.
    __builtin_prefetch((const char*)kslice + lane * 216, 0, 1);

    v8f acc = {};
    int kc = 0;                     // global K base of current chunk-of-4
#pragma unroll
    for (int di = 0; di < 3; ++di) {
#pragma unroll
        for (int dj = 0; dj < 3; ++dj) {
#pragma unroll
            for (int dk = 0; dk < 3; ++dk) {
                const float* xv = xv_base + di * XS_D + dj * XS_H + dk * XS_W;
#pragma unroll
                for (int cc = 0; cc < 8; cc += 4) {
                    // A 16x4 f32: VGPR0 = K {0 | 2}, VGPR1 = K {1 | 3} by half
                    // -> per lane two adjacent channels: one b64 load
                    v2f a = *(const v2f*)(xv + cc);

                    // B 4x16 f32: VGPR0 = row 2*half, VGPR1 = row 2*half+1
                    v2f b;
                    const int krow = kc + 2 * half;
                    b.x = kp[(size_t)krow * F_OUT];
                    b.y = kp[(size_t)(krow + 1) * F_OUT];

                    acc = __builtin_amdgcn_wmma_f32_16x16x4_f32(
                        /*neg_a=*/false, a, /*neg_b=*/false, b,
                        /*c_mod=*/(short)0, acc,
                        /*reuse_a=*/false, /*reuse_b=*/false);
                    kc += 4;
                }
            }
        }
    }

    // D 16x16 f32: VGPR r, lanes 0-15 -> M=r, N=lane.
    // acc[0] lanes 0..7 = out[b=0, p, f],  acc[1] lanes 0..7 = out[b=1, p, f]
    const float o0 = acc[0];
    const float o1 = acc[1];
    if (lane < 8) {
        const float bv = bias[(size_t)p * F_OUT + lane];
        out[(size_t)p * F_OUT + lane]                          = o0 + bv;
        out[(size_t)P_TOTAL * F_OUT + (size_t)p * F_OUT + lane] = o1 + bv;
    }
}

extern "C" void kernel_launch(void* const* d_in, const int* in_sizes, int n_in,
                              void* d_out, int out_size, void* d_ws, size_t ws_size,
                              hipStream_t stream) {
    const float* x    = (const float*)d_in[0];   // (2,24,24,24,8)
    const float* kern = (const float*)d_in[1];   // (10648,216,8)
    const float* bias = (const float*)d_in[2];   // (22,22,22,8)
    float* out        = (float*)d_out;           // (2,22,22,22,8)

    (void)in_sizes; (void)n_in; (void)out_size; (void)d_ws; (void)ws_size;

    // 10648 sites, one wave32 per site, 8 waves (256 threads) per block.
    dim3 grid(P_TOTAL / 8);   // 1331, exact
    dim3 block(256);
    lc3d_wmma_f32_kernel<<<grid, block, 0, stream>>>(x, kern, bias, out);
}